// category_process_25099788878134
// MI455X (gfx1250) — compile-verified
//
#include <hip/hip_runtime.h>
#include <stdint.h>

// Problem constants (fixed by the reference).
#define B_ 4096
#define L_ 200
#define V_ 100000
#define D_ 300

#define ROWS_PER_CHUNK 8                       // TDM gather, 32-bit idx -> max 8 rows/op
#define NCHUNK (L_ / ROWS_PER_CHUNK)           // 25
#define CHUNK_FLOATS (ROWS_PER_CHUNK * D_)     // 2400 floats
#define CHUNK_BYTES (CHUNK_FLOATS * 4)         // 9600 bytes
#define NBUF 4                                 // quad buffer, issue-depth 2

typedef unsigned int v4u __attribute__((ext_vector_type(4)));
typedef int          v4i __attribute__((ext_vector_type(4)));
typedef int          v8i __attribute__((ext_vector_type(8)));

// Issue one TDM gather: 8 rows of D_ floats, indexed by idp[0..7], into LDS at
// byte offset lds_byte. Descriptor per CDNA5 ISA.
__device__ __forceinline__ void tdm_gather8(const float* __restrict__ table,
                                            const int* __restrict__ idp,
                                            unsigned lds_byte) {
  unsigned long long ga = (unsigned long long)(uintptr_t)table;

  // D# group 0: count=1 (valid), gather_index_size=1 (32-bit), gather_mode=1,
  // lds_addr, global_addr (57b), type=2 ("image").
  v4u g0;
  g0.x = 0xC0000001u;
  g0.y = lds_byte;
  g0.z = (unsigned)(ga & 0xFFFFFFFFull);
  g0.w = ((unsigned)(ga >> 32) & 0x01FFFFFFu) | (2u << 30);

  // D# group 1: workgroup_mask=0 (not in cluster), data_size=2 (4 bytes),
  // tensor_dim0=D_, tensor_dim1=V_ (OOB bound), tile_dim0=D_,
  // tile_dim1=#indices=8, tensor_dim0_stride=D_ elements.
  v8i g1;
  g1[0] = (int)(2u << 16);                                           // data_size=4B
  g1[1] = (int)(((unsigned)D_ & 0xFFFFu) << 16);                     // dim0 lo16
  g1[2] = (int)((((unsigned)D_ >> 16) & 0xFFFFu) |
                (((unsigned)V_ & 0xFFFFu) << 16));                   // dim0 hi | dim1 lo
  g1[3] = (int)((((unsigned)V_ >> 16) & 0xFFFFu) |
                (((unsigned)D_ & 0xFFFFu) << 16));                   // dim1 hi | tile_dim0
  g1[4] = ROWS_PER_CHUNK;                                            // tile_dim1 = #idx
  g1[5] = D_;                                                        // dim0 stride lo32
  g1[6] = 0;
  g1[7] = 0;

  // D# groups 2/3 (gather mode, 32-bit): row_index_0..7.
  v4i g2, g3;
  g2[0] = idp[0]; g2[1] = idp[1]; g2[2] = idp[2]; g2[3] = idp[3];
  g3[0] = idp[4]; g3[1] = idp[5]; g3[2] = idp[6]; g3[3] = idp[7];

#if defined(__clang_major__) && (__clang_major__ >= 23)
  v8i z8 = {0, 0, 0, 0, 0, 0, 0, 0};
  __builtin_amdgcn_tensor_load_to_lds(g0, g1, g2, g3, z8, 0);
#else
  __builtin_amdgcn_tensor_load_to_lds(g0, g1, g2, g3, 0);
#endif
}

__global__ __launch_bounds__(256)
void category_embbag_kernel(const int* __restrict__ ids,
                            const float* __restrict__ emb,
                            float* __restrict__ out) {
  // LDS: quad-buffered gather stage (4 x 2400 floats) + 8 wave counts.
  // Partial sums reuse buffer 0 after the gather loop.
  __shared__ __align__(16) float smem[NBUF * CHUNK_FLOATS + 8];

  const int b    = blockIdx.x;
  const int tid  = threadIdx.x;
  const int w    = tid >> 5;     // wave id 0..7
  const int lane = tid & 31;

  const int* idrow = ids + (size_t)b * L_;

  // LDS byte offset of smem (low 32 bits of the generic LDS pointer).
  const unsigned smem_base = (unsigned)(uintptr_t)(&smem[0]);

  // Scalar (wave-uniform) predicate so only wave 0 issues TDM ops.
  const bool issuer = (__builtin_amdgcn_readfirstlane(tid) < 32);

  if (issuer) {
    tdm_gather8(emb, idrow, smem_base);                                // chunk 0
    tdm_gather8(emb, idrow + ROWS_PER_CHUNK, smem_base + CHUNK_BYTES); // chunk 1
  }

  float4 a0 = make_float4(0.f, 0.f, 0.f, 0.f);
  float4 a1 = a0;
  float4 a2 = a0;
  int wcnt = 0;

  for (int c = 0; c < NCHUNK; ++c) {
    if (issuer) {
      // Issue 2 ahead; buf[(c+2)&3] held chunk c-2, which every wave finished
      // reading before passing barrier A(c-1), which wave0 already crossed.
      if (c + 2 < NCHUNK) {
        tdm_gather8(emb, idrow + (c + 2) * ROWS_PER_CHUNK,
                    smem_base + (unsigned)(((c + 2) & (NBUF - 1)) * CHUNK_BYTES));
        __builtin_amdgcn_s_wait_tensorcnt(2);   // oldest (chunk c) done
      } else if (c + 1 < NCHUNK) {
        __builtin_amdgcn_s_wait_tensorcnt(1);
      } else {
        __builtin_amdgcn_s_wait_tensorcnt(0);
      }
    }
    __syncthreads();   // A(c): chunk c visible in LDS to all waves

    // Wave w reduces row w of this chunk: 75 float4 chunks across 32 lanes.
    const float* row = smem + (c & (NBUF - 1)) * CHUNK_FLOATS + w * D_;
    float4 v0 = *(const float4*)(row + (lane      ) * 4);
    float4 v1 = *(const float4*)(row + (lane + 32) * 4);
    float4 v2 = make_float4(0.f, 0.f, 0.f, 0.f);
    if (lane < (75 - 64)) v2 = *(const float4*)(row + (lane + 64) * 4);

    a0.x += v0.x; a0.y += v0.y; a0.z += v0.z; a0.w += v0.w;
    a1.x += v1.x; a1.y += v1.y; a1.z += v1.z; a1.w += v1.w;
    a2.x += v2.x; a2.y += v2.y; a2.z += v2.z; a2.w += v2.w;

    bool nz = (v0.x != 0.f) | (v0.y != 0.f) | (v0.z != 0.f) | (v0.w != 0.f) |
              (v1.x != 0.f) | (v1.y != 0.f) | (v1.z != 0.f) | (v1.w != 0.f) |
              (v2.x != 0.f) | (v2.y != 0.f) | (v2.z != 0.f) | (v2.w != 0.f);
    if (__ballot(nz) != 0ull) wcnt++;   // wave-uniform keep count
  }

  __syncthreads();   // all waves done reading the last chunk (lives in buf[0])

  // Deterministic cross-wave reduction through LDS (reuse buffer 0).
  float* part = smem;
  float* cnt  = smem + NBUF * CHUNK_FLOATS;
  *(float4*)(part + w * D_ + (lane      ) * 4) = a0;
  *(float4*)(part + w * D_ + (lane + 32) * 4) = a1;
  if (lane < (75 - 64)) *(float4*)(part + w * D_ + (lane + 64) * 4) = a2;
  if (lane == 0) cnt[w] = (float)wcnt;
  __syncthreads();

  float cs = 0.f;
#pragma unroll
  for (int i = 0; i < 8; ++i) cs += cnt[i];

  {
    int d = tid;                         // 0..255 (< D_)
    float s = 0.f;
#pragma unroll
    for (int i = 0; i < 8; ++i) s += part[i * D_ + d];
    out[(size_t)b * D_ + d] = (cs > 0.f) ? (s / cs) : 0.f;
  }
  if (tid < (D_ - 256)) {                // dims 256..299
    int d = 256 + tid;
    float s = 0.f;
#pragma unroll
    for (int i = 0; i < 8; ++i) s += part[i * D_ + d];
    out[(size_t)b * D_ + d] = (cs > 0.f) ? (s / cs) : 0.f;
  }
}

extern "C" void kernel_launch(void* const* d_in, const int* in_sizes, int n_in,
                              void* d_out, int out_size, void* d_ws, size_t ws_size,
                              hipStream_t stream) {
  const int*   ids = (const int*)d_in[0];    // [B, L] int32
  const float* emb = (const float*)d_in[1];  // [V, D] float32
  float*       out = (float*)d_out;          // [B, D] float32
  (void)in_sizes; (void)n_in; (void)out_size; (void)d_ws; (void)ws_size;

  dim3 grid(B_);
  dim3 block(256);
  hipLaunchKernelGGL(category_embbag_kernel, grid, block, 0, stream, ids, emb, out);
}